// TemporalGNN_36773509988866
// MI455X (gfx1250) — compile-verified
//
#include <hip/hip_runtime.h>
#include <math.h>

#define FEAT 8
#define OUT 64
#define PERIODS 12

typedef __attribute__((ext_vector_type(2))) float v2f;
typedef __attribute__((ext_vector_type(8))) float v8f;

// ---------------------------------------------------------------------------
// Graph prep kernels
// ---------------------------------------------------------------------------
__global__ void fill_zero_k(float* __restrict__ p, int n) {
  int i = blockIdx.x * blockDim.x + threadIdx.x;
  if (i < n) p[i] = 0.0f;
}

__global__ void deg_k(const int* __restrict__ ei, float* deg, int E) {
  int e = blockIdx.x * blockDim.x + threadIdx.x;
  if (e < E) atomicAdd(&deg[ei[E + e]], 1.0f);
}

__global__ void dinv_k(float* dinv, int N) {
  int n = blockIdx.x * blockDim.x + threadIdx.x;
  if (n < N) dinv[n] = rsqrtf(dinv[n] + 1.0f);
}

__global__ void norme_k(const int* __restrict__ ei, const float* __restrict__ dinv,
                        float* __restrict__ norme, int E) {
  int e = blockIdx.x * blockDim.x + threadIdx.x;
  if (e < E) norme[e] = dinv[ei[e]] * dinv[ei[E + e]];
}

__global__ void softmax12_k(const float* __restrict__ att, float* __restrict__ probs) {
  if (threadIdx.x == 0 && blockIdx.x == 0) {
    float m = att[0];
    for (int i = 1; i < PERIODS; i++) m = fmaxf(m, att[i]);
    float e[PERIODS];
    float s = 0.0f;
    for (int i = 0; i < PERIODS; i++) { e[i] = __expf(att[i] - m); s += e[i]; }
    for (int i = 0; i < PERIODS; i++) probs[i] = e[i] / s;
  }
}

// ---------------------------------------------------------------------------
// GCN: xw = x[:,:,t] @ W  (K=8, tiny -> VALU)
// ---------------------------------------------------------------------------
__global__ void xw_k(const float* __restrict__ x, const float* __restrict__ W,
                     float* __restrict__ xw, int t, int total) {
  int i = blockIdx.x * blockDim.x + threadIdx.x;
  if (i >= total) return;
  int n = i >> 6, o = i & 63;
  const float* xr = x + (long long)n * (FEAT * PERIODS) + t;
  float s = 0.0f;
#pragma unroll
  for (int f = 0; f < FEAT; f++) s = fmaf(xr[f * PERIODS], W[f * OUT + o], s);
  xw[i] = s;
}

// agg = xw * dinv^2 (self loop) + bias
__global__ void agg_init_k(const float* __restrict__ xw, const float* __restrict__ dinv,
                           const float* __restrict__ b, float* __restrict__ agg, int total) {
  int i = blockIdx.x * blockDim.x + threadIdx.x;
  if (i >= total) return;
  int n = i >> 6, o = i & 63;
  float dv = dinv[n];
  agg[i] = fmaf(xw[i], dv * dv, b[o]);
}

// agg[dst] += xw[src] * norm_e
// One thread per (edge, 4-feature chunk): one b128 load of xw, 4 f32 atomics.
__global__ void edge_scatter_k(const int* __restrict__ ei, const float* __restrict__ xw,
                               const float* __restrict__ norme, float* agg, int E) {
  int i = blockIdx.x * blockDim.x + threadIdx.x;
  if (i >= (E << 4)) return;
  int e = i >> 4, q = (i & 15) << 2;
  int s = ei[e];
  int d = ei[E + e];
  float w = norme[e];
  const float4 v = *(const float4*)(xw + (s << 6) + q);
  float* dst = agg + (d << 6) + q;
  atomicAdd(dst + 0, v.x * w);
  atomicAdd(dst + 1, v.y * w);
  atomicAdd(dst + 2, v.z * w);
  atomicAdd(dst + 3, v.w * w);
}

// ---------------------------------------------------------------------------
// WMMA f32 16x16x4 with LDS-staged B.
// LDS B layout: K-row pairs adjacent per column so each lane reads its
// {B[k][n], B[k+1][n]} as a single aligned ds_load_b64:
//   Bs[(k>>1)*128 + n*2 + (k&1)] = B[k][n]
// One wave computes a full 16x64 output row-block: 4 col-tiles, 4 v8f accs,
// each A v2f loaded once and reused by 4 WMMAs.
// ---------------------------------------------------------------------------
__device__ __forceinline__ void tile_mm128_lds(const float* __restrict__ A0,
                                               const float* __restrict__ A1,
                                               const float* __restrict__ Bs,
                                               int node0, v8f acc[4]) {
  const int lane = threadIdx.x & 31;
  const int m = lane & 15;                 // A row / B col within tile
  const int k0 = (lane >> 4) << 1;         // 0 or 2 (upper half-wave K shift)
  const float* Ar0 = A0 + ((node0 + m) << 6);
  const float* Ar1 = A1 + ((node0 + m) << 6);
#pragma unroll
  for (int kk = 0; kk < 64; kk += 4) {
    v2f a;
    a.x = Ar0[kk + k0];
    a.y = Ar0[kk + k0 + 1];
    const float* bp = Bs + ((kk + k0) >> 1) * 128 + (m << 1);
#pragma unroll
    for (int c = 0; c < 4; c++) {
      v2f b = *(const v2f*)(bp + (c << 5));   // col n = c*16 + m
      acc[c] = __builtin_amdgcn_wmma_f32_16x16x4_f32(false, a, false, b, (short)0,
                                                     acc[c], false, false);
    }
  }
#pragma unroll
  for (int kk = 0; kk < 64; kk += 4) {
    v2f a;
    a.x = Ar1[kk + k0];
    a.y = Ar1[kk + k0 + 1];
    const float* bp = Bs + (((64 + kk + k0) >> 1) * 128) + (m << 1);
#pragma unroll
    for (int c = 0; c < 4; c++) {
      v2f b = *(const v2f*)(bp + (c << 5));
      acc[c] = __builtin_amdgcn_wmma_f32_16x16x4_f32(false, a, false, b, (short)0,
                                                     acc[c], false, false);
    }
  }
}

// cooperative stage of a 128x64 weight panel into paired LDS layout
__device__ __forceinline__ void stage_B(float* __restrict__ Bs,
                                        const float* __restrict__ LW) {
#pragma unroll
  for (int j = 0; j < 32; j++) {
    int idx = (j << 8) + threadIdx.x;      // 32 * 256 = 8192 floats
    int k = idx >> 6, n = idx & 63;
    Bs[(k >> 1) * 128 + (n << 1) + (k & 1)] = LW[idx];
  }
}

// Z = sigmoid([aggZ|H] @ LzW + Lzb); R = sigmoid([aggR|H] @ LrW + Lrb); HR = H*R
// blockIdx.y = gate (0=Z, 1=R); 8 waves/block, one node tile per wave.
__global__ __launch_bounds__(256) void gates_zr_k(
    const float* __restrict__ aggZ, const float* __restrict__ aggR,
    const float* __restrict__ H,
    const float* __restrict__ LzW, const float* __restrict__ Lzb,
    const float* __restrict__ LrW, const float* __restrict__ Lrb,
    float* __restrict__ Zbuf, float* __restrict__ HRbuf, int ntiles) {
  __shared__ float Bs[8192];               // 32 KB of the 320 KB/WGP LDS
  const int gate = blockIdx.y;
  const float* A0 = (gate == 0) ? aggZ : aggR;
  const float* LW = (gate == 0) ? LzW : LrW;
  const float* Lb = (gate == 0) ? Lzb : Lrb;

  stage_B(Bs, LW);
  __syncthreads();

  const int tile = blockIdx.x * 8 + (threadIdx.x >> 5);
  if (tile >= ntiles) return;              // wave-uniform; after barrier
  const int node0 = tile << 4;

  v8f acc[4] = {{0.f}, {0.f}, {0.f}, {0.f}};
  tile_mm128_lds(A0, H, Bs, node0, acc);

  const int lane = threadIdx.x & 31;
  const int mbase = (lane >> 4) << 3;      // 0 or 8
  const int mcol = lane & 15;
#pragma unroll
  for (int c = 0; c < 4; c++) {
    const int ncol = (c << 4) + mcol;
    const float bv = Lb[ncol];
#pragma unroll
    for (int v = 0; v < 8; v++) {
      int row = node0 + mbase + v;
      float g = 1.0f / (1.0f + __expf(-(acc[c][v] + bv)));
      if (gate == 0) Zbuf[(row << 6) + ncol] = g;
      else           HRbuf[(row << 6) + ncol] = H[(row << 6) + ncol] * g;
    }
  }
}

// Ht = tanh([aggH|HR] @ LhW + Lhb); Hn = Z*H + (1-Z)*Ht; H=Hn; Hacc += p_t*Hn
__global__ __launch_bounds__(256) void gate_h_k(
    const float* __restrict__ aggH, const float* __restrict__ HR,
    const float* __restrict__ Zbuf,
    const float* __restrict__ LhW, const float* __restrict__ Lhb,
    float* __restrict__ H, float* __restrict__ Hacc,
    const float* __restrict__ probs, int t, int ntiles) {
  __shared__ float Bs[8192];

  stage_B(Bs, LhW);
  __syncthreads();

  const int tile = blockIdx.x * 8 + (threadIdx.x >> 5);
  if (tile >= ntiles) return;
  const int node0 = tile << 4;

  v8f acc[4] = {{0.f}, {0.f}, {0.f}, {0.f}};
  tile_mm128_lds(aggH, HR, Bs, node0, acc);

  const int lane = threadIdx.x & 31;
  const int mbase = (lane >> 4) << 3;
  const int mcol = lane & 15;
  const float p = probs[t];
#pragma unroll
  for (int c = 0; c < 4; c++) {
    const int ncol = (c << 4) + mcol;
    const float bv = Lhb[ncol];
#pragma unroll
    for (int v = 0; v < 8; v++) {
      int row = node0 + mbase + v;
      int idx = (row << 6) + ncol;
      float ht = tanhf(acc[c][v] + bv);
      float z = Zbuf[idx];
      float h = H[idx];
      float hn = z * h + (1.0f - z) * ht;
      H[idx] = hn;
      Hacc[idx] = fmaf(p, hn, Hacc[idx]);
    }
  }
}

// out = relu(Hacc) @ head_W + head_b   ([N,64]@[64,12])
__global__ void head_k(const float* __restrict__ Hacc, const float* __restrict__ hW,
                       const float* __restrict__ hb, float* __restrict__ out, int total) {
  int i = blockIdx.x * blockDim.x + threadIdx.x;
  if (i >= total) return;
  int n = i / PERIODS, p = i % PERIODS;
  float s = hb[p];
  const float* hr = Hacc + (n << 6);
#pragma unroll 8
  for (int o = 0; o < OUT; o++) s = fmaf(fmaxf(hr[o], 0.0f), hW[o * PERIODS + p], s);
  out[i] = s;
}

// ---------------------------------------------------------------------------
extern "C" void kernel_launch(void* const* d_in, const int* in_sizes, int n_in,
                              void* d_out, int out_size, void* d_ws, size_t ws_size,
                              hipStream_t stream) {
  const float* x    = (const float*)d_in[0];
  const int*   ei   = (const int*)d_in[1];
  const float* Wz   = (const float*)d_in[2];
  const float* bz   = (const float*)d_in[3];
  const float* LzW  = (const float*)d_in[4];
  const float* Lzb  = (const float*)d_in[5];
  const float* Wr   = (const float*)d_in[6];
  const float* br   = (const float*)d_in[7];
  const float* LrW  = (const float*)d_in[8];
  const float* Lrb  = (const float*)d_in[9];
  const float* Wh   = (const float*)d_in[10];
  const float* bh   = (const float*)d_in[11];
  const float* LhW  = (const float*)d_in[12];
  const float* Lhb  = (const float*)d_in[13];
  const float* att  = (const float*)d_in[14];
  const float* hW   = (const float*)d_in[15];
  const float* hb   = (const float*)d_in[16];
  float* out = (float*)d_out;

  const int N = in_sizes[0] / (FEAT * PERIODS);   // 20000
  const int E = in_sizes[1] / 2;                  // 640000
  const int NO = N * OUT;
  const int ntiles = N / 16;                      // 1250 (exact)

  // workspace layout (floats)
  float* ws    = (float*)d_ws;
  float* dinv  = ws;                 // N (also used as deg accumulator)
  float* probs = dinv + N;           // 16
  float* norme = probs + 16;         // E
  float* xw    = norme + E;          // N*OUT
  float* aggZ  = xw   + NO;
  float* aggR  = aggZ + NO;
  float* aggH  = aggR + NO;
  float* H     = aggH + NO;          // H and Hacc contiguous (one zero-fill)
  float* Hacc  = H    + NO;
  float* Zbuf  = Hacc + NO;
  float* HRbuf = Zbuf + NO;

  const int T = 256;
  auto cdiv = [](int a, int b) { return (a + b - 1) / b; };

  // graph prep + state init
  fill_zero_k<<<cdiv(N, T), T, 0, stream>>>(dinv, N);
  fill_zero_k<<<cdiv(2 * NO, T), T, 0, stream>>>(H, 2 * NO);
  deg_k<<<cdiv(E, T), T, 0, stream>>>(ei, dinv, E);
  dinv_k<<<cdiv(N, T), T, 0, stream>>>(dinv, N);
  norme_k<<<cdiv(E, T), T, 0, stream>>>(ei, dinv, norme, E);
  softmax12_k<<<1, 32, 0, stream>>>(att, probs);

  const float* Ws[3]   = {Wz, Wr, Wh};
  const float* bs[3]   = {bz, br, bh};
  float*       aggs[3] = {aggZ, aggR, aggH};

  const int gblocks = cdiv(ntiles, 8);            // 8 node tiles per 256-thr block

  for (int t = 0; t < PERIODS; t++) {
    for (int g = 0; g < 3; g++) {
      xw_k<<<cdiv(NO, T), T, 0, stream>>>(x, Ws[g], xw, t, NO);
      agg_init_k<<<cdiv(NO, T), T, 0, stream>>>(xw, dinv, bs[g], aggs[g], NO);
      edge_scatter_k<<<cdiv(E * 16, T), T, 0, stream>>>(ei, xw, norme, aggs[g], E);
    }
    gates_zr_k<<<dim3(gblocks, 2), 256, 0, stream>>>(aggZ, aggR, H, LzW, Lzb, LrW, Lrb,
                                                     Zbuf, HRbuf, ntiles);
    gate_h_k<<<gblocks, 256, 0, stream>>>(aggH, HRbuf, Zbuf, LhW, Lhb,
                                          H, Hacc, probs, t, ntiles);
  }

  head_k<<<cdiv(N * PERIODS, T), T, 0, stream>>>(Hacc, hW, hb, out, N * PERIODS);
}